// vacf_27968827031657
// MI455X (gfx1250) — compile-verified
//
#include <hip/hip_runtime.h>

// ---------------------------------------------------------------------------
// VACF via banded Gram matrix on CDNA5 (gfx1250).
//   vel: [2000][12000] f32.  vacf[t] = sum_i vel[i].vel[i+t] / ((2000-t)*12000)
// Only tile-pairs (I, I+d), d=0..7 cover lags 0..99 (each (i,i+t) pair lands
// in exactly one 16x16 tile pair).  fp32 WMMA 16x16x4 matches the reference
// GEMM numerics.  Double-buffered GLOBAL_LOAD_ASYNC_TO_LDS_B128 staging
// (ASYNCcnt) overlaps the L2->LDS copy of stage s+1 with the WMMAs of stage s;
// one workgroup barrier per stage.
// ---------------------------------------------------------------------------

typedef float v2f __attribute__((ext_vector_type(2)));
typedef float v8f __attribute__((ext_vector_type(8)));

#define T_TOTAL    2000
#define M_TOTAL    12000
#define NLAGS      100
#define NTILES     125         // 2000 / 16
#define NDIAG      8           // d = 0..7: max lag in pair = 16*7+15 = 127 >= 99
#define KC         480         // k-chunk per stage; 12000/480 = 25 stages
#define NSTAGES    25
#define LDS_STRIDE 484         // 484 % 64 == 36, gcd(36,64)=4 -> 16 distinct
                               // bank groups of 4 across rows: conflict-free
#define TILE_FLOATS (16 * LDS_STRIDE)            // 7744 floats per tile
#define BUF_FLOATS  (2 * TILE_FLOATS)            // A tile | B tile per buffer
#define LDS_TOTAL_BYTES (2 * BUF_FLOATS * 4)     // double buffered: 123,904 B

__global__ void vacf_zero_ws(float* ws) {
    if (threadIdx.x < NLAGS) ws[threadIdx.x] = 0.0f;
}

// Issue one stage's A+B tiles as async LDS copies:
// 2 tiles * 16 rows * 120 float4 = 3840 b128 ops == 15 per thread.
__device__ __forceinline__
void issue_stage(unsigned long long velU, unsigned ldsBase,
                 int tid, int i0, int j0, int kbase, int buf) {
    #pragma unroll
    for (int j = 0; j < 15; ++j) {
        const int u    = tid + 256 * j;
        const int tile = (u >= 1920) ? 1 : 0;
        const int rem  = u - tile * 1920;
        const int row  = rem / 120;
        const int c4   = rem - row * 120;
        const int grow = (tile ? j0 : i0) + row;
        const unsigned gOff = (unsigned)(((unsigned)grow * M_TOTAL + kbase + (c4 << 2)) * 4u);
        const unsigned lOff = ldsBase +
            (unsigned)((buf * BUF_FLOATS + tile * TILE_FLOATS
                        + row * LDS_STRIDE + (c4 << 2)) * 4);
        // GVS mode: 64-bit SGPR base + per-lane 32-bit byte offset, b128 payload.
        asm volatile("global_load_async_to_lds_b128 %0, %1, %2 offset:0"
                     :: "v"(lOff), "v"(gOff), "s"(velU)
                     : "memory");
    }
}

__global__ __launch_bounds__(256)
void vacf_band_gemm(const float* __restrict__ vel, float* __restrict__ ws) {
    extern __shared__ float lds[];               // 2 buffers of (A|B) tiles

    const int I = blockIdx.x;                    // 0..124
    const int d = blockIdx.y;                    // 0..7
    const int J = I + d;
    if (J >= NTILES) return;                     // uniform: barrier-safe

    const int i0 = I * 16;
    const int j0 = J * 16;

    const int tid  = threadIdx.x;
    const int lane = tid & 31;
    const int wave = tid >> 5;

    // WMMA f32 16x16x4 fragment addressing (ISA 7.12.2):
    //   lanes 0-15 : row = lane,    VGPR0/1 = K+0,K+1
    //   lanes 16-31: row = lane-16, VGPR0/1 = K+2,K+3
    const int frow = lane & 15;
    const int koff = (lane >> 4) << 1;

    const unsigned long long velU = (unsigned long long)vel;
    // Generic shared pointer -> 32-bit LDS byte offset (aperture: addr[31:0]).
    const unsigned ldsBase = (unsigned)(unsigned long long)(void*)lds;

    v8f acc0, acc1;
    #pragma unroll
    for (int i = 0; i < 8; ++i) { acc0[i] = 0.0f; acc1[i] = 0.0f; }

    // Prologue: fill buffer 0 with stage 0.
    issue_stage(velU, ldsBase, tid, i0, j0, 0, 0);

    for (int s = 0; s < NSTAGES; ++s) {
        // Drain this thread's copies for stage s, then make all waves' copies
        // (and all stage s-1 consumers) globally ordered with one barrier.
        asm volatile("s_wait_asynccnt 0" ::: "memory");
        __syncthreads();

        // Prefetch stage s+1 into the other buffer; overlaps with WMMAs below.
        if (s + 1 < NSTAGES)
            issue_stage(velU, ldsBase, tid, i0, j0, (s + 1) * KC, (s + 1) & 1);

        // Each wave reduces its private 60-wide k slice: 15 WMMAs,
        // split over two accumulators to break the C->D RAW chain.
        const float* aF = lds + (s & 1) * BUF_FLOATS + frow * LDS_STRIDE + koff;
        const float* bF = aF + TILE_FLOATS;
        int kk = wave * 60;
        #pragma unroll
        for (int stp = 0; stp < 15; ++stp) {
            v2f a = *(const v2f*)(aF + kk);
            v2f b = *(const v2f*)(bF + kk);
            if (stp & 1)
                acc1 = __builtin_amdgcn_wmma_f32_16x16x4_f32(
                           false, a, false, b, (short)0, acc1, false, false);
            else
                acc0 = __builtin_amdgcn_wmma_f32_16x16x4_f32(
                           false, a, false, b, (short)0, acc0, false, false);
            kk += 4;
        }
    }

    // ---- Cross-wave reduction of the 8 partial G tiles through LDS.
    __syncthreads();
    {
        const int base = wave * 256 + lane * 8;
        #pragma unroll
        for (int i = 0; i < 8; ++i) lds[base + i] = acc0[i] + acc1[i];
    }
    __syncthreads();

    {
        float sum = 0.0f;
        #pragma unroll
        for (int w = 0; w < 8; ++w) sum += lds[w * 256 + tid];

        // Decode (M,N) from C/D layout: flat = lane*8 + v;
        //   M = v + 8*(lane>>4), N = lane & 15.
        const int cl = tid >> 3;
        const int v  = tid & 7;
        const int Mr = v + ((cl >> 4) << 3);
        const int Nc = cl & 15;
        const int t  = (j0 + Nc) - (i0 + Mr);
        if (t >= 0 && t < NLAGS) atomicAdd(&ws[t], sum);
    }
}

__global__ void vacf_finalize(const float* __restrict__ ws, float* __restrict__ out) {
    const int t = threadIdx.x;
    if (t < NLAGS) {
        const float denom = (float)(T_TOTAL - t) * (float)M_TOTAL;
        out[t] = ws[t] / denom;
    }
}

extern "C" void kernel_launch(void* const* d_in, const int* in_sizes, int n_in,
                              void* d_out, int out_size, void* d_ws, size_t ws_size,
                              hipStream_t stream) {
    const float* vel = (const float*)d_in[0];   // [2000*4000*3] f32
    float*       out = (float*)d_out;           // [100] f32
    float*       ws  = (float*)d_ws;            // >= 400 B scratch accumulator

    vacf_zero_ws<<<1, 128, 0, stream>>>(ws);

    dim3 grid(NTILES, NDIAG);
    vacf_band_gemm<<<grid, 256, LDS_TOTAL_BYTES, stream>>>(vel, ws);

    vacf_finalize<<<1, 128, 0, stream>>>(ws, out);
}